// TSM_25864293057205
// MI455X (gfx1250) — compile-verified
//
#include <hip/hip_runtime.h>
#include <hip/hip_bf16.h>
#include <math.h>

// ---------------------------------------------------------------------------
// softmax(-cdist^2(x,x)/T) for x:[4,4096,128] f32 -> out:[4,4096,4096,1] f32
// Store-bound (268 MB output @ 23.3 TB/s ~= 11.5us floor); x (8MB) is L2-resident.
// GEMM uses exact-precision V_WMMA_F32_16X16X4_F32; full softmax row staged in
// the 320KB WGP LDS so the Gram matrix is computed in a single pass.
// ---------------------------------------------------------------------------

typedef float v2f __attribute__((ext_vector_type(2)));
typedef float v8f __attribute__((ext_vector_type(8)));

#define BATCH      4
#define SEQ        4096
#define DIM        128
#define TEMP       13.544f
#define INV_TEMP   (1.0f / 13.544f)

// LDS layout (bytes)
#define SOUT_STRIDE 4100                       // floats; 4100%64!=0 -> no bank conflicts
#define SOUT_BYTES  (16 * SOUT_STRIDE * 4)     // 262400
#define SA_STRIDE   132                        // floats; conflict-free b64 fragment reads
#define SA_BYTES    (16 * SA_STRIDE * 4)       // 8448
#define SSQ_BYTES   (SEQ * 4)                  // 16384
#define SMEM_BYTES  (SOUT_BYTES + SA_BYTES + SSQ_BYTES)  // 287232 < 320KB

// ---------------------------------------------------------------------------
// Kernel 1: row squared norms. One wave32 per row; lane loads one float4.
// ---------------------------------------------------------------------------
__global__ void TSM_sqnorm_kernel(const float* __restrict__ x,
                                  float* __restrict__ sq) {
    const int wave = (int)((blockIdx.x * blockDim.x + threadIdx.x) >> 5);
    const int lane = (int)(threadIdx.x & 31u);
    if (wave >= BATCH * SEQ) return;  // uniform per wave

    const float4 v = *(const float4*)(x + (size_t)wave * DIM + lane * 4);
    float s = v.x * v.x + v.y * v.y + v.z * v.z + v.w * v.w;
#pragma unroll
    for (int m = 16; m >= 1; m >>= 1) s += __shfl_xor(s, m, 32);
    if (lane == 0) sq[wave] = s;
}

// ---------------------------------------------------------------------------
// Kernel 2: one workgroup (8 waves) per 16-row tile.
//   Phase 1: GEMM epilogue -> LDS row buffer (pre-softmax logits)
//   Phase 2: row softmax in LDS, coalesced float4 store to global
// ---------------------------------------------------------------------------
__global__ __launch_bounds__(256) void TSM_cdist_softmax_kernel(
    const float* __restrict__ x,
    const float* __restrict__ sq,
    float* __restrict__ out) {

    extern __shared__ char smem_raw[];
    float* sOut = (float*)smem_raw;                          // 16 x 4100 f32
    float* sA   = (float*)(smem_raw + SOUT_BYTES);           // 16 x 132  f32
    float* sSq  = (float*)(smem_raw + SOUT_BYTES + SA_BYTES);// 4096 f32

    const int tid    = (int)threadIdx.x;
    const int lane   = tid & 31;
    const int waveId = tid >> 5;          // 0..7
    const int half   = lane >> 4;         // 0 or 1 (wave32 half)
    const int l16    = lane & 15;

    const int b       = (int)blockIdx.x / (SEQ / 16);
    const int rowTile = (int)blockIdx.x % (SEQ / 16);
    const int m0      = rowTile * 16;

    const float* xb = x + (size_t)b * SEQ * DIM;

    // --- Stage A tile (16 rows x 128) into LDS, padded stride 132 ----------
    {
        const float* src = xb + (size_t)m0 * DIM;
        for (int idx = tid; idx < 16 * (DIM / 4); idx += 256) {
            const int r  = idx >> 5;       // row 0..15
            const int c4 = idx & 31;       // float4 col
            const float4 v = *(const float4*)(src + r * DIM + c4 * 4);
            *(float4*)(sA + r * SA_STRIDE + c4 * 4) = v;
        }
        // stage all squared norms of this batch (re-read many times)
        for (int i = tid; i < SEQ; i += 256) sSq[i] = sq[b * SEQ + i];
    }
    __syncthreads();

    // --- Load persistent A operand: 32 k-steps x v2f (64 VGPRs) -----------
    // A 16x4 f32 layout: VGPR0 = K=k (lanes 0-15) / K=k+2 (lanes 16-31),
    //                    VGPR1 = K=k+1 / K=k+3  -> one b64 LDS read per lane.
    v2f afrag[DIM / 4];
#pragma unroll
    for (int kk = 0; kk < DIM / 4; ++kk) {
        afrag[kk] = *(const v2f*)(sA + l16 * SA_STRIDE + 4 * kk + 2 * half);
    }

    // --- Phase 1: sweep column tiles (each wave owns 32 of 256) ------------
    for (int tIdx = waveId; tIdx < SEQ / 16; tIdx += 8) {
        const int n0 = tIdx * 16;

        // B 4x16 f32 fragments: same lane pattern as A, read from global (L2)
        v2f bfrag[DIM / 4];
#pragma unroll
        for (int kk = 0; kk < DIM / 4; ++kk) {
            bfrag[kk] = *(const v2f*)(xb + (size_t)(n0 + l16) * DIM +
                                      4 * kk + 2 * half);
        }

        // two accumulator chains to cut the serial WMMA RAW dependency
        v8f acc0 = {};
        v8f acc1 = {};
#pragma unroll
        for (int kk = 0; kk < DIM / 4; kk += 2) {
            acc0 = __builtin_amdgcn_wmma_f32_16x16x4_f32(
                false, afrag[kk],     false, bfrag[kk],
                (short)0, acc0, false, false);
            acc1 = __builtin_amdgcn_wmma_f32_16x16x4_f32(
                false, afrag[kk + 1], false, bfrag[kk + 1],
                (short)0, acc1, false, false);
        }

        // epilogue: logits = -max(|a|^2 + |b|^2 - 2 a.b, 0) / T  -> LDS
        // C/D layout: VGPR r -> M = r + 8*half, N = lane&15
        const float sb = sSq[n0 + l16];
#pragma unroll
        for (int r = 0; r < 8; ++r) {
            const int   M   = r + 8 * half;
            const float dot = acc0[r] + acc1[r];
            const float sa  = sSq[m0 + M];
            float d2 = sa + sb - 2.0f * dot;
            d2 = fmaxf(d2, 0.0f);
            sOut[M * SOUT_STRIDE + n0 + l16] = -d2 * INV_TEMP;
        }
    }
    __syncthreads();

    // --- Phase 2: row softmax + store (each wave owns 2 rows) --------------
#pragma unroll
    for (int rr = 0; rr < 2; ++rr) {
        const int row = waveId * 2 + rr;
        float* srow = sOut + row * SOUT_STRIDE;

        // row max
        float mx = -INFINITY;
        for (int i = lane; i < SEQ; i += 32) mx = fmaxf(mx, srow[i]);
#pragma unroll
        for (int m = 16; m >= 1; m >>= 1) mx = fmaxf(mx, __shfl_xor(mx, m, 32));

        // exp + sum (overwrite LDS with exponentials)
        float sum = 0.0f;
        for (int i = lane; i < SEQ; i += 32) {
            const float e = __expf(srow[i] - mx);
            srow[i] = e;
            sum += e;
        }
#pragma unroll
        for (int m = 16; m >= 1; m >>= 1) sum += __shfl_xor(sum, m, 32);
        const float inv = 1.0f / sum;

        // normalized coalesced float4 store
        float* orow = out + ((size_t)(b * SEQ + m0 + row)) * SEQ;
        for (int i4 = lane; i4 < SEQ / 4; i4 += 32) {
            float4 v = *(const float4*)(srow + i4 * 4);
            v.x *= inv; v.y *= inv; v.z *= inv; v.w *= inv;
            *(float4*)(orow + i4 * 4) = v;
        }
    }
}

// ---------------------------------------------------------------------------
extern "C" void kernel_launch(void* const* d_in, const int* in_sizes, int n_in,
                              void* d_out, int out_size, void* d_ws, size_t ws_size,
                              hipStream_t stream) {
    const float* x   = (const float*)d_in[0];
    float*       out = (float*)d_out;
    float*       sq  = (float*)d_ws;   // 4*4096 f32 = 64 KB scratch

    (void)in_sizes; (void)n_in; (void)out_size; (void)ws_size;

    static bool attr_set = false;   // host-side launch config only (idempotent)
    if (!attr_set) {
        hipFuncSetAttribute((const void*)TSM_cdist_softmax_kernel,
                            hipFuncAttributeMaxDynamicSharedMemorySize,
                            SMEM_BYTES);
        attr_set = true;
    }

    // Kernel 1: squared norms (16384 rows, 8 waves/block)
    TSM_sqnorm_kernel<<<(BATCH * SEQ) / 8, 256, 0, stream>>>(x, sq);

    // Kernel 2: 1024 workgroups (4 batches x 256 row tiles), 287 KB LDS each
    TSM_cdist_softmax_kernel<<<BATCH * (SEQ / 16), 256, SMEM_BYTES, stream>>>(
        x, sq, out);
}